// EOTLayer_70815420776813
// MI455X (gfx1250) — compile-verified
//
#include <hip/hip_runtime.h>
#include <math.h>

// Problem dims (fixed by the reference).
#define HH   512
#define WW   512
#define CC   3
#define NN   32
#define HW   (HH * WW)            // 262144 pixels
#define HWC  (HW * CC)            // 786432 elements per image
#define TPB  256                  // 8 wave32s per block
#define PIX_BLOCKS (HW / TPB)     // 1024 blocks per transform
#define NPART (PIX_BLOCKS * NN)   // 32768 block partial sums

// ---------------------------------------------------------------------------
// Fast transcendentals for gfx1250.
//  - tanh: prefer hardware V_TANH_F32 (single TRANS op, co-executes with the
//    main VALU pipe). Fallback: exp2+rcp formulation (~2 TRANS + 4 VALU)
//    instead of the ~20-op libm expansion.
//  - rcp:  V_RCP_F32 (1 ulp) instead of the ~10-op IEEE division sequence.
//    In this workload denom==1.0 exactly (c0=c1=0), where rcp is exact.
// ---------------------------------------------------------------------------
__device__ __forceinline__ float fast_rcp(float x) {
#if __has_builtin(__builtin_amdgcn_rcpf)
    return __builtin_amdgcn_rcpf(x);
#else
    return 1.0f / x;
#endif
}

__device__ __forceinline__ float fast_tanh(float x) {
#if __has_builtin(__builtin_amdgcn_tanhf)
    return __builtin_amdgcn_tanhf(x);
#else
    // tanh(x) = (e^{2x}-1)/(e^{2x}+1); clamp to keep e^{2x} finite.
    float xc = fminf(fmaxf(x, -15.0f), 15.0f);
#if __has_builtin(__builtin_amdgcn_exp2f)
    float e = __builtin_amdgcn_exp2f(xc * 2.8853900817779268f); // 2/ln(2)
#else
    float e = exp2f(xc * 2.8853900817779268f);
#endif
    return (e - 1.0f) * fast_rcp(e + 1.0f);
#endif
}

// ---------------------------------------------------------------------------
// Kernel A: one-time precompute of atanh(image) and mask*inter (3 MB each).
// Only 786K elements -> keep the precise libm atanhf here.
// ---------------------------------------------------------------------------
__global__ __launch_bounds__(TPB)
void eot_precompute(const float* __restrict__ image,
                    const float* __restrict__ mask,
                    const float* __restrict__ inter,
                    float* __restrict__ wimg,
                    float* __restrict__ msrc) {
    int i = blockIdx.x * TPB + threadIdx.x;
    if (i < HWC) {
        wimg[i] = atanhf(image[i]);
        msrc[i] = mask[i] * inter[i];
    }
}

// ---------------------------------------------------------------------------
// Kernel B: per-(transform, pixel) projective bilinear warp + tanh + store,
// with per-block partial sum of (image - constrained)^2.
//
//   grid = (NN, PIX_BLOCKS): n = blockIdx.x (fast-varying) so the 32
//   transforms of the SAME pixel tile dispatch back-to-back -> their nearly
//   identical source reads (taps / image / wimg) hit hot L0/L2 lines instead
//   of being re-fetched once per full image sweep. n is uniform per block ->
//   transform coefficients become scalar loads.
//
// PRECOMP=false fallback recomputes atanh / mask*inter inline if ws is tiny.
// ---------------------------------------------------------------------------
template<bool PRECOMP>
__global__ __launch_bounds__(TPB)
void eot_warp(const float* __restrict__ image,
              const float* __restrict__ mask,
              const float* __restrict__ inter,
              const float* __restrict__ transforms,
              const float* __restrict__ wimg,
              const float* __restrict__ msrc,
              float* __restrict__ out,
              float* __restrict__ partials) {
    const int n   = blockIdx.x;
    const int pix = blockIdx.y * TPB + threadIdx.x;   // always < HW
    const int x   = pix & (WW - 1);
    const int y   = pix >> 9;                         // log2(WW)

    // Uniform across the block -> scalar loads.
    const float a0 = transforms[n * 8 + 0];
    const float a1 = transforms[n * 8 + 1];
    const float a2 = transforms[n * 8 + 2];
    const float b0 = transforms[n * 8 + 3];
    const float b1 = transforms[n * 8 + 4];
    const float b2 = transforms[n * 8 + 5];
    const float c0 = transforms[n * 8 + 6];
    const float c1 = transforms[n * 8 + 7];

    const float fx = (float)x, fy = (float)y;
    const float inv = fast_rcp(c0 * fx + c1 * fy + 1.0f);
    const float sx  = (a0 * fx + a1 * fy + a2) * inv;
    const float sy  = (b0 * fx + b1 * fy + b2) * inv;

    const float x0f = floorf(sx), y0f = floorf(sy);
    const float wx = sx - x0f,   wy = sy - y0f;
    const float x1f = x0f + 1.0f, y1f = y0f + 1.0f;

    const float WM = (float)(WW - 1), HM = (float)(HH - 1);
    const float vx0 = (x0f >= 0.0f && x0f <= WM) ? 1.0f : 0.0f;
    const float vx1 = (x1f >= 0.0f && x1f <= WM) ? 1.0f : 0.0f;
    const float vy0 = (y0f >= 0.0f && y0f <= HM) ? 1.0f : 0.0f;
    const float vy1 = (y1f >= 0.0f && y1f <= HM) ? 1.0f : 0.0f;

    const int ix0 = (int)fminf(fmaxf(x0f, 0.0f), WM);
    const int ix1 = (int)fminf(fmaxf(x1f, 0.0f), WM);
    const int iy0 = (int)fminf(fmaxf(y0f, 0.0f), HM);
    const int iy1 = (int)fminf(fmaxf(y1f, 0.0f), HM);

    const int base00 = (iy0 * WW + ix0) * CC;
    const int base01 = (iy0 * WW + ix1) * CC;
    const int base10 = (iy1 * WW + ix0) * CC;
    const int base11 = (iy1 * WW + ix1) * CC;

    const float m00 = vx0 * vy0, m01 = vx1 * vy0;
    const float m10 = vx0 * vy1, m11 = vx1 * vy1;

    const int    pc      = pix * CC;
    const size_t outbase = (size_t)n * HWC + (size_t)pc;
    float lsum = 0.0f;

#pragma unroll
    for (int c = 0; c < CC; ++c) {
        float v00, v01, v10, v11;
        if (PRECOMP) {
            v00 = msrc[base00 + c]; v01 = msrc[base01 + c];
            v10 = msrc[base10 + c]; v11 = msrc[base11 + c];
        } else {
            v00 = mask[base00 + c] * inter[base00 + c];
            v01 = mask[base01 + c] * inter[base01 + c];
            v10 = mask[base10 + c] * inter[base10 + c];
            v11 = mask[base11 + c] * inter[base11 + c];
        }
        v00 *= m00; v01 *= m01; v10 *= m10; v11 *= m11;

        const float top    = v00 + wx * (v01 - v00);
        const float bot    = v10 + wx * (v11 - v10);
        const float warped = top + wy * (bot - top);

        const float wim = PRECOMP ? wimg[pc + c] : atanhf(image[pc + c]);
        const float con = fast_tanh(wim + warped);

        // 100 MB streamed output, never re-read: non-temporal store keeps the
        // hot 9 MB of source data resident in L2 (TH=NT on global_store).
        __builtin_nontemporal_store(con, &out[outbase + c]);

        const float pert = image[pc + c] - con;
        lsum = fmaf(pert, pert, lsum);
    }

    // wave32 reduction, then cross-wave via LDS (8 waves per block).
#pragma unroll
    for (int off = 16; off > 0; off >>= 1)
        lsum += __shfl_xor(lsum, off, 32);

    __shared__ float smem[TPB / 32];
    const int lane = threadIdx.x & 31;
    const int wave = threadIdx.x >> 5;
    if (lane == 0) smem[wave] = lsum;
    __syncthreads();
    if (threadIdx.x < (TPB / 32)) {
        float v = smem[threadIdx.x];
#pragma unroll
        for (int off = (TPB / 64); off > 0; off >>= 1)
            v += __shfl_xor(v, off, 32);
        if (threadIdx.x == 0)
            partials[blockIdx.x * PIX_BLOCKS + blockIdx.y] = v;
    }
}

// ---------------------------------------------------------------------------
// Kernel C: deterministic final reduction of 32768 partials (double accum),
// writes loss = 0.01 * sqrt(sum). Single block -> fixed summation order.
// ---------------------------------------------------------------------------
__global__ __launch_bounds__(TPB)
void eot_finalize(const float* __restrict__ partials,
                  float* __restrict__ loss_out) {
    double s = 0.0;
    for (int i = threadIdx.x; i < NPART; i += TPB)
        s += (double)partials[i];

#pragma unroll
    for (int off = 16; off > 0; off >>= 1)
        s += __shfl_xor(s, off, 32);

    __shared__ double smem[TPB / 32];
    const int lane = threadIdx.x & 31;
    const int wave = threadIdx.x >> 5;
    if (lane == 0) smem[wave] = s;
    __syncthreads();
    if (threadIdx.x < (TPB / 32)) {
        double v = smem[threadIdx.x];
#pragma unroll
        for (int off = (TPB / 64); off > 0; off >>= 1)
            v += __shfl_xor(v, off, 32);
        if (threadIdx.x == 0) {
            // Smoothness term is identically 0 (empty batch slice in reference).
            loss_out[0] = 0.01f * sqrtf((float)v);
        }
    }
}

// ---------------------------------------------------------------------------
extern "C" void kernel_launch(void* const* d_in, const int* in_sizes, int n_in,
                              void* d_out, int out_size, void* d_ws, size_t ws_size,
                              hipStream_t stream) {
    const float* image      = (const float*)d_in[0];   // [1,H,W,C]
    const float* mask       = (const float*)d_in[1];   // [1,H,W,C]
    const float* inter      = (const float*)d_in[2];   // [1,H,W,C]
    const float* transforms = (const float*)d_in[3];   // [N,8]

    float* out      = (float*)d_out;                   // [N,H,W,C] + loss
    float* loss_out = out + (size_t)NN * HWC;          // last element

    const size_t need_full = (size_t)HWC * 4 * 2 + (size_t)NPART * 4;

    if (ws_size >= need_full) {
        float* wimg     = (float*)d_ws;          // atanh(image), 3 MB
        float* msrc     = wimg + HWC;            // mask*inter,  3 MB
        float* partials = msrc + HWC;            // 128 KB

        eot_precompute<<<(HWC + TPB - 1) / TPB, TPB, 0, stream>>>(
            image, mask, inter, wimg, msrc);
        eot_warp<true><<<dim3(NN, PIX_BLOCKS), TPB, 0, stream>>>(
            image, mask, inter, transforms, wimg, msrc, out, partials);
        eot_finalize<<<1, TPB, 0, stream>>>(partials, loss_out);
    } else {
        // Fallback: tiny workspace; recompute atanh / mask*inter inline.
        float* partials = (float*)d_ws;          // needs 128 KB
        eot_warp<false><<<dim3(NN, PIX_BLOCKS), TPB, 0, stream>>>(
            image, mask, inter, transforms, nullptr, nullptr, out, partials);
        eot_finalize<<<1, TPB, 0, stream>>>(partials, loss_out);
    }
}